// HetGTAN_NoSem_76682346102826
// MI455X (gfx1250) — compile-verified
//
#include <hip/hip_runtime.h>

#define HOP   3
#define NA    50000
#define NP    100000
#define NIN   256
#define NHID  128
#define NOUT  64
#define NEDGE 500000
#define SCAN_B 512

typedef __attribute__((ext_vector_type(2))) float v2f;
typedef __attribute__((ext_vector_type(8))) float v8f;

__device__ __forceinline__ float wave_sum32(float v) {
#pragma unroll
  for (int off = 16; off > 0; off >>= 1)
    v += __shfl_xor(v, off, 32);
  return v;
}

// ---------------------------------------------------------------------------
// C[M x N] = act(X[M x K] @ W[K x N] + bias), fp32 WMMA 16x16x4.
// One wave computes a 16(M) x 64(N) tile; A-fragment reused across 4 N-tiles.
// M multiple of 16, N multiple of 64.
// ---------------------------------------------------------------------------
template <int N, int K, bool RELU>
__launch_bounds__(32)
__global__ void gemm_wmma_f32(const float* __restrict__ X,
                              const float* __restrict__ W,
                              const float* __restrict__ bias,
                              float* __restrict__ out) {
  const int lane = threadIdx.x;        // 0..31
  const int m0   = blockIdx.x << 4;
  const int n0   = blockIdx.y << 6;
  const int row  = m0 + (lane & 15);
  const int kh   = lane >> 4;          // 0 or 1 -> K-pair select
  const int col  = lane & 15;

  v8f acc[4] = {};
  const float* xrow = X + (size_t)row * K + 2 * kh;

  for (int k = 0; k < K; k += 4) {
    v2f a = *(const v2f*)(xrow + k);
    const float* wp = W + (size_t)(k + 2 * kh) * N + n0 + col;
#pragma unroll
    for (int nt = 0; nt < 4; ++nt) {
      v2f b;
      b.x = wp[nt * 16];
      b.y = wp[nt * 16 + N];
      acc[nt] = __builtin_amdgcn_wmma_f32_16x16x4_f32(
          false, a, false, b, (short)0, acc[nt], false, false);
    }
  }

#pragma unroll
  for (int nt = 0; nt < 4; ++nt) {
#pragma unroll
    for (int r = 0; r < 8; ++r) {
      const int rr = m0 + r + (kh << 3);
      const int cc = n0 + (nt << 4) + col;
      float v = acc[nt][r] + bias[cc];
      if (RELU) v = fmaxf(v, 0.0f);
      out[(size_t)rr * N + cc] = v;
    }
  }
}

// ---------------------------------------------------------------------------
// Per-node attention scalars (wave per node, float4 loads, shuffle reduce)
// ---------------------------------------------------------------------------
__launch_bounds__(256)
__global__ void src_scores(const float* __restrict__ x,
                           const float* __restrict__ a1,
                           const float* __restrict__ a2,
                           float* __restrict__ s1,
                           float* __restrict__ selfw, int n) {
  const int gid  = blockIdx.x * blockDim.x + threadIdx.x;
  const int v    = gid >> 5;
  const int lane = gid & 31;
  if (v >= n) return;
  const float4 xv  = ((const float4*)(x + (size_t)v * NHID))[lane];
  const float4 a1v = ((const float4*)a1)[lane];
  const float4 a2v = ((const float4*)a2)[lane];
  float d1 = xv.x * a1v.x + xv.y * a1v.y + xv.z * a1v.z + xv.w * a1v.w;
  float d2 = xv.x * a2v.x + xv.y * a2v.y + xv.z * a2v.z + xv.w * a2v.w;
  d1 = wave_sum32(d1);
  d2 = wave_sum32(d2);
  if (lane == 0) {
    s1[v] = d1;
    float z = d1 + d2;
    z = z > 0.0f ? z : 0.2f * z;
    selfw[v] = expf(z);
  }
}

__launch_bounds__(256)
__global__ void tgt_scores(const float* __restrict__ h,
                           const float* __restrict__ a2,
                           float* __restrict__ t2, int n) {
  const int gid  = blockIdx.x * blockDim.x + threadIdx.x;
  const int v    = gid >> 5;
  const int lane = gid & 31;
  if (v >= n) return;
  const float4 hv  = ((const float4*)(h + (size_t)v * NHID))[lane];
  const float4 a2v = ((const float4*)a2)[lane];
  float d = hv.x * a2v.x + hv.y * a2v.y + hv.z * a2v.z + hv.w * a2v.w;
  d = wave_sum32(d);
  if (lane == 0) t2[v] = d;
}

// ---------------------------------------------------------------------------
// CSR build (once per launch; edge lists are hop-invariant)
// ---------------------------------------------------------------------------
__launch_bounds__(256)
__global__ void zero_i32(int* __restrict__ p, int n) {
  const int i = blockIdx.x * blockDim.x + threadIdx.x;
  if (i < n) p[i] = 0;
}

__launch_bounds__(256)
__global__ void edge_count(const int* __restrict__ s, int* __restrict__ cnt,
                           int nE) {
  const int e = blockIdx.x * blockDim.x + threadIdx.x;
  if (e < nE) atomicAdd(&cnt[s[e]], 1);
}

// Inclusive Hillis-Steele scan per 512-block; emits per-block totals.
__launch_bounds__(SCAN_B)
__global__ void scan_block(const int* __restrict__ in, int* __restrict__ out,
                           int* __restrict__ bsum, int n) {
  __shared__ int sm[SCAN_B];
  const int tid = threadIdx.x;
  const int gid = blockIdx.x * SCAN_B + tid;
  sm[tid] = (gid < n) ? in[gid] : 0;
  __syncthreads();
  for (int off = 1; off < SCAN_B; off <<= 1) {
    int t = (tid >= off) ? sm[tid - off] : 0;
    __syncthreads();
    sm[tid] += t;
    __syncthreads();
  }
  if (gid < n) out[gid] = sm[tid];
  if (tid == SCAN_B - 1) bsum[blockIdx.x] = sm[SCAN_B - 1];
}

__launch_bounds__(SCAN_B)
__global__ void scan_add(int* __restrict__ out, const int* __restrict__ bscan,
                         int n) {
  const int gid = blockIdx.x * SCAN_B + threadIdx.x;
  if (gid < n && blockIdx.x > 0) out[gid] += bscan[blockIdx.x - 1];
}

__launch_bounds__(256)
__global__ void cursor_init(int* __restrict__ cur, const int* __restrict__ csum,
                            int n) {
  const int v = blockIdx.x * blockDim.x + threadIdx.x;
  if (v < n) cur[v] = v ? csum[v - 1] : 0;
}

__launch_bounds__(256)
__global__ void csr_fill(const int* __restrict__ s, const int* __restrict__ t,
                         int* __restrict__ cur, int* __restrict__ tperm,
                         int nE) {
  const int e = blockIdx.x * blockDim.x + threadIdx.x;
  if (e >= nE) return;
  const int pos = atomicAdd(&cur[s[e]], 1);
  tperm[pos] = t[e];
}

// ---------------------------------------------------------------------------
// Fused per-hop aggregation: one wave per source node, gather-only, no
// atomics. acc = selfw*x[v] + sum_j w_j * h[tperm[j]];  wsum = selfw + sum w_j
// out = elu(acc / wsum).
// ---------------------------------------------------------------------------
__launch_bounds__(256)
__global__ void node_agg(const int* __restrict__ csum,
                         const int* __restrict__ tperm,
                         const float* __restrict__ s1,
                         const float* __restrict__ selfw,
                         const float* __restrict__ t2,
                         const float* __restrict__ h,
                         const float* __restrict__ x,
                         float* __restrict__ hout, int n) {
  const int gid  = blockIdx.x * blockDim.x + threadIdx.x;
  const int v    = gid >> 5;
  const int lane = gid & 31;
  if (v >= n) return;
  const int start = v ? csum[v - 1] : 0;
  const int end   = csum[v];
  const float s1v = s1[v];
  const float sw  = selfw[v];

  const float4 xv = ((const float4*)(x + (size_t)v * NHID))[lane];
  float4 acc;
  acc.x = sw * xv.x; acc.y = sw * xv.y; acc.z = sw * xv.z; acc.w = sw * xv.w;
  float wsum = sw;

  for (int j = start; j < end; ++j) {
    const int tv = tperm[j];
    float z = s1v + t2[tv];
    z = z > 0.0f ? z : 0.2f * z;
    const float w = expf(z);
    wsum += w;
    const float4 hv = ((const float4*)(h + (size_t)tv * NHID))[lane];
    acc.x += w * hv.x; acc.y += w * hv.y; acc.z += w * hv.z; acc.w += w * hv.w;
  }

  const float inv = 1.0f / wsum;
  float4 o;
  float z0 = acc.x * inv; o.x = z0 > 0.0f ? z0 : expm1f(z0);
  float z1 = acc.y * inv; o.y = z1 > 0.0f ? z1 : expm1f(z1);
  float z2 = acc.z * inv; o.z = z2 > 0.0f ? z2 : expm1f(z2);
  float z3 = acc.w * inv; o.w = z3 > 0.0f ? z3 : expm1f(z3);
  ((float4*)(hout + (size_t)v * NHID))[lane] = o;
}

// ---------------------------------------------------------------------------
extern "C" void kernel_launch(void* const* d_in, const int* in_sizes, int n_in,
                              void* d_out, int out_size, void* d_ws,
                              size_t ws_size, hipStream_t stream) {
  (void)in_sizes; (void)n_in; (void)out_size; (void)ws_size;

  const float* x_A   = (const float*)d_in[0];
  const float* x_P   = (const float*)d_in[1];
  const int*   sAP   = (const int*)d_in[2];
  const int*   tAP   = (const int*)d_in[3];
  const int*   sPA   = (const int*)d_in[4];
  const int*   tPA   = (const int*)d_in[5];
  const float* W1_A  = (const float*)d_in[6];
  const float* b1_A  = (const float*)d_in[7];
  const float* W1_P  = (const float*)d_in[8];
  const float* b1_P  = (const float*)d_in[9];
  const float* attn1 = (const float*)d_in[10];
  const float* attn2 = (const float*)d_in[11];
  const float* W2    = (const float*)d_in[12];
  const float* b2    = (const float*)d_in[13];
  float* out = (float*)d_out;

  // ---- workspace layout ----
  float* p   = (float*)d_ws;
  float* xAp = p; p += (size_t)NA * NHID;
  float* xPp = p; p += (size_t)NP * NHID;
  float* hA0 = p; p += (size_t)NA * NHID;
  float* hA1 = p; p += (size_t)NA * NHID;
  float* hP0 = p; p += (size_t)NP * NHID;
  float* hP1 = p; p += (size_t)NP * NHID;
  float* s1A   = p; p += NA;
  float* selfA = p; p += NA;
  float* t2A   = p; p += NA;
  float* s1P   = p; p += NP;
  float* selfP = p; p += NP;
  float* t2P   = p; p += NP;
  int* ip      = (int*)p;
  int* csumA   = ip; ip += NA;     // inclusive degree prefix sum (A sources)
  int* curA    = ip; ip += NA;     // counts, then fill cursors
  int* tpermAP = ip; ip += NEDGE;  // targets grouped by A source
  int* csumP   = ip; ip += NP;
  int* curP    = ip; ip += NP;
  int* tpermPA = ip; ip += NEDGE;
  int* bs      = ip; ip += SCAN_B; // block sums
  int* bs2     = ip; ip += SCAN_B; // scanned block sums
  int* bsDummy = ip; ip += 4;

  const int nblkA = (NA + SCAN_B - 1) / SCAN_B;   // 98
  const int nblkP = (NP + SCAN_B - 1) / SCAN_B;   // 196
  const dim3 edgeGrid((NEDGE + 255) / 256);

  // ---- CSR build for A-side segments (edges sAP -> tAP) ----
  zero_i32<<<(NA + 255) / 256, 256, 0, stream>>>(curA, NA);
  edge_count<<<edgeGrid, 256, 0, stream>>>(sAP, curA, NEDGE);
  scan_block<<<nblkA, SCAN_B, 0, stream>>>(curA, csumA, bs, NA);
  scan_block<<<1, SCAN_B, 0, stream>>>(bs, bs2, bsDummy, nblkA);
  scan_add<<<nblkA, SCAN_B, 0, stream>>>(csumA, bs2, NA);
  cursor_init<<<(NA + 255) / 256, 256, 0, stream>>>(curA, csumA, NA);
  csr_fill<<<edgeGrid, 256, 0, stream>>>(sAP, tAP, curA, tpermAP, NEDGE);

  // ---- CSR build for P-side segments (edges sPA -> tPA) ----
  zero_i32<<<(NP + 255) / 256, 256, 0, stream>>>(curP, NP);
  edge_count<<<edgeGrid, 256, 0, stream>>>(sPA, curP, NEDGE);
  scan_block<<<nblkP, SCAN_B, 0, stream>>>(curP, csumP, bs, NP);
  scan_block<<<1, SCAN_B, 0, stream>>>(bs, bs2, bsDummy, nblkP);
  scan_add<<<nblkP, SCAN_B, 0, stream>>>(csumP, bs2, NP);
  cursor_init<<<(NP + 255) / 256, 256, 0, stream>>>(curP, csumP, NP);
  csr_fill<<<edgeGrid, 256, 0, stream>>>(sPA, tPA, curP, tpermPA, NEDGE);

  // ---- input projections (fp32 WMMA) ----
  gemm_wmma_f32<NHID, NIN, true>
      <<<dim3(NA / 16, NHID / 64), 32, 0, stream>>>(x_A, W1_A, b1_A, xAp);
  gemm_wmma_f32<NHID, NIN, true>
      <<<dim3(NP / 16, NHID / 64), 32, 0, stream>>>(x_P, W1_P, b1_P, xPp);

  const float* curAh = xAp;  // hA starts as xA
  const float* curPh = xPp;  // hP starts as xP
  float* bufA[2] = {hA0, hA1};
  float* bufP[2] = {hP0, hP1};

  const dim3 nodeBlkA((NA + 7) / 8), nodeBlkP((NP + 7) / 8);

  for (int i = 0; i < HOP; ++i) {
    const float* a1A = attn1 + (size_t)(i * 2 + 0) * NHID;
    const float* a2A = attn2 + (size_t)(i * 2 + 0) * NHID;
    const float* a1P = attn1 + (size_t)(i * 2 + 1) * NHID;
    const float* a2P = attn2 + (size_t)(i * 2 + 1) * NHID;
    float* nxtA = bufA[i & 1];
    float* nxtP = bufP[i & 1];

    src_scores<<<nodeBlkA, 256, 0, stream>>>(xAp, a1A, a2A, s1A, selfA, NA);
    src_scores<<<nodeBlkP, 256, 0, stream>>>(xPp, a1P, a2P, s1P, selfP, NP);
    tgt_scores<<<nodeBlkP, 256, 0, stream>>>(curPh, a2A, t2P, NP);
    tgt_scores<<<nodeBlkA, 256, 0, stream>>>(curAh, a2P, t2A, NA);

    // gather-only fused aggregation + self term + normalize + ELU
    node_agg<<<nodeBlkA, 256, 0, stream>>>(csumA, tpermAP, s1A, selfA, t2P,
                                           curPh, xAp, nxtA, NA);
    node_agg<<<nodeBlkP, 256, 0, stream>>>(csumP, tpermPA, s1P, selfP, t2A,
                                           curAh, xPp, nxtP, NP);

    curAh = nxtA;
    curPh = nxtP;
  }

  // ---- output projection: out = hA @ W2 + b2 ----
  gemm_wmma_f32<NOUT, NHID, false>
      <<<dim3(NA / 16, 1), 32, 0, stream>>>(curAh, W2, b2, out);
}